// LlamaAttention_88287347737180
// MI455X (gfx1250) — compile-verified
//
#include <hip/hip_runtime.h>

#define B_DIM  2
#define S_DIM  1024
#define HIDDEN 2048
#define NH     32
#define NKV    8
#define HD     64

typedef float v2f __attribute__((ext_vector_type(2)));
typedef float v8f __attribute__((ext_vector_type(8)));
typedef unsigned int v4u __attribute__((ext_vector_type(4)));
typedef int v8i __attribute__((ext_vector_type(8)));
typedef int v4i __attribute__((ext_vector_type(4)));

static __device__ __forceinline__ v2f ld2(const float* p) {
    return *reinterpret_cast<const v2f*>(p);
}

// ---------------------------------------------------------------------------
// TDM: DMA a 2D f32 tile (tile_cols x tile_rows) from global into LDS.
// D# per CDNA5 ISA: group0 = {flags, lds_addr, global_addr,
// type=2}; group1 = {data_size=4B, tensor dims, tile dims, row stride};
// groups 2/3 zero (<=2D tensor). Tracked with TENSORcnt.
// This toolchain exposes the 6-arg builtin:
//   (uint32x4 g0, int32x8 g1, int32x4, int32x4, int32x8, i32 cpol)
// ---------------------------------------------------------------------------
static __device__ __forceinline__ void tdm_load_2d(const float* gsrc,
                                                   const float* lds_dst,
                                                   unsigned tile_cols,
                                                   unsigned tile_rows,
                                                   unsigned tensor_cols,
                                                   unsigned tensor_rows,
                                                   unsigned long long row_stride)
{
    unsigned long long ga = (unsigned long long)(uintptr_t)gsrc;
    unsigned lds_off = (unsigned)(uintptr_t)lds_dst;   // LDS aperture: addr[31:0]
    v4u g0;
    g0[0] = 1u;                                   // count=1, user-mode load
    g0[1] = lds_off;                              // lds_addr
    g0[2] = (unsigned)ga;                         // global_addr[31:0]
    g0[3] = ((unsigned)(ga >> 32) & 0x01FFFFFFu) | (2u << 30);  // [56:32] | type=2
    v8i g1;
    g1[0] = (int)(2u << 16);                      // wg_mask=0, data_size=4B
    g1[1] = (int)((tensor_cols & 0xFFFFu) << 16); // tensor_dim0[15:0]
    g1[2] = (int)((tensor_cols >> 16) | ((tensor_rows & 0xFFFFu) << 16));
    g1[3] = (int)((tensor_rows >> 16) | (tile_cols << 16));     // tile_dim0
    g1[4] = (int)tile_rows;                       // tile_dim1 (tile_dim2=0)
    g1[5] = (int)(unsigned)row_stride;            // tensor_dim0_stride[31:0]
    g1[6] = (int)(unsigned)((row_stride >> 32) & 0xFFFFull);
    g1[7] = 0;
    v4i z4 = (v4i)0;
    v8i z8 = (v8i)0;
    __builtin_amdgcn_tensor_load_to_lds(g0, g1, z4, z4, z8, 0);
}

// ---------------------------------------------------------------------------
// Generic 16x16-tile f32 GEMM using V_WMMA_F32_16X16X4_F32. One wave per tile.
// W tiles double-buffered into LDS via TDM; A rows loaded as coalesced float2.
// head_major!=0 stores out as [h][b][s][d] (h = col/64), else row-major MxN.
// Dynamic LDS: 2 * 16*16 floats = 2048 B.
// ---------------------------------------------------------------------------
__global__ __launch_bounds__(32)
void gemm16_wmma(const float* __restrict__ X, const float* __restrict__ W,
                 float* __restrict__ out, int M, int N, int K, int head_major)
{
    extern __shared__ float wsm[];                 // [2][16*16]
    int lane  = threadIdx.x;
    int m0    = blockIdx.y * 16;
    int n0    = blockIdx.x * 16;
    int mrow  = lane & 15;
    int nl    = lane & 15;
    int koff  = (lane >> 4) << 1;   // lanes 0-15: K=0,1 ; lanes 16-31: K=2,3
    int rbase = (lane >> 4) * 8;    // C rows: lanes 0-15 -> M=r, 16-31 -> M=r+8

    const float* arow = X + (size_t)(m0 + mrow) * K;
    v8f c = {};

    auto step = [&](int kb, const float* wt) {
        __builtin_prefetch(arow + kb * 16 + 256, 0, 0);   // global_prefetch_b8
#pragma unroll
        for (int j = 0; j < 4; ++j) {
            int kl = j * 4 + koff;
            v2f a = ld2(arow + kb * 16 + kl);
            v2f b;
            b.x = wt[kl * 16 + nl];
            b.y = wt[(kl + 1) * 16 + nl];
            c = __builtin_amdgcn_wmma_f32_16x16x4_f32(false, a, false, b,
                                                      (short)0, c, false, false);
        }
    };

    const int NB = K >> 4;
    tdm_load_2d(W + (size_t)0 * N + n0, wsm, 16, 16, (unsigned)N, (unsigned)K,
                (unsigned long long)N);
    for (int kb = 0; kb < NB - 1; ++kb) {
        tdm_load_2d(W + (size_t)(kb + 1) * 16 * N + n0, wsm + ((kb + 1) & 1) * 256,
                    16, 16, (unsigned)N, (unsigned)K, (unsigned long long)N);
        __builtin_amdgcn_s_wait_tensorcnt(1);
        step(kb, wsm + (kb & 1) * 256);
    }
    __builtin_amdgcn_s_wait_tensorcnt(0);
    step(NB - 1, wsm + ((NB - 1) & 1) * 256);

#pragma unroll
    for (int r = 0; r < 8; ++r) {
        int row = m0 + rbase + r;
        int col = n0 + nl;
        size_t idx;
        if (head_major) {
            int bb = row >> 10;          // row / S_DIM
            int ss = row & (S_DIM - 1);
            int hh = col >> 6;
            int dd = col & 63;
            idx = (((size_t)hh * B_DIM + bb) * S_DIM + ss) * HD + dd;
        } else {
            idx = (size_t)row * N + col;
        }
        out[idx] = c[r];
    }
}

// ---------------------------------------------------------------------------
// In-place RoPE on head-major [h][b][s][d] buffer. One thread per (d,d+32)
// pair -> no read/write race.
// ---------------------------------------------------------------------------
__global__ void rope_kernel(float* __restrict__ qk, const float* __restrict__ cosT,
                            const float* __restrict__ sinT, int nHeads)
{
    int id = blockIdx.x * blockDim.x + threadIdx.x;
    int total = nHeads * B_DIM * S_DIM * (HD / 2);
    if (id >= total) return;
    int d  = id & 31;
    int s  = (id >> 5) & (S_DIM - 1);
    int hb = id >> 15;                       // h*B + b
    float* base = qk + ((size_t)hb * S_DIM + s) * HD;
    float x0 = base[d];
    float x1 = base[d + 32];
    float c0 = cosT[s * HD + d],      s0 = sinT[s * HD + d];
    float c1 = cosT[s * HD + d + 32], s1 = sinT[s * HD + d + 32];
    base[d]      = x0 * c0 - x1 * s0;        // x*cos + (-x_hi)*sin
    base[d + 32] = x1 * c1 + x0 * s1;        // x*cos + ( x_lo)*sin
}

// ---------------------------------------------------------------------------
// Attention: one wave per (b, h, 16-row query tile).
// Dynamic LDS layout (73728 B):
//   [0,   65536)  : score strip  [16][S]
//   [65536, 69632): tile slot 0  (16x64 f32)  - K tiles / V tiles via TDM
//   [69632, 73728): tile slot 1
// Phase 1: scores = (Q Kt)/8 + causal/pad mask (K tiles TDM double-buffered)
// Phase 2: wave-local softmax; probs -> global attn out (+ zero tail)
// Phase 3: ctx = P @ V, kt-outer with 4 accumulators (V tiles TDM d-buffered)
// ---------------------------------------------------------------------------
__global__ __launch_bounds__(32)
void attn_kernel(const float* __restrict__ Q, const float* __restrict__ Kv,
                 const float* __restrict__ Vv, const float* __restrict__ amask,
                 float* __restrict__ attn_out, float* __restrict__ ctx)
{
    extern __shared__ float smem[];
    float* sc    = smem;                       // 16 * S_DIM floats
    float* tile0 = smem + 16 * S_DIM;          // 16 * HD floats
    float* tile1 = tile0 + 16 * HD;

    int qt = blockIdx.x, h = blockIdx.y, b = blockIdx.z;
    int lane  = threadIdx.x;
    int hkv   = h >> 2;                        // GROUP = 4
    int q0    = qt * 16;
    int mrow  = lane & 15;
    int nl    = lane & 15;
    int koff  = (lane >> 4) << 1;
    int rbase = (lane >> 4) * 8;

    const float* Qb = Q  + (((size_t)h   * B_DIM + b) * S_DIM) * HD;
    const float* Kb = Kv + (((size_t)hkv * B_DIM + b) * S_DIM) * HD;
    const float* Vb = Vv + (((size_t)hkv * B_DIM + b) * S_DIM) * HD;

    // Q-row fragments: loaded once, reused for every K tile.
    const float* arow = Qb + (size_t)(q0 + mrow) * HD;
    v2f aq[16];
#pragma unroll
    for (int i = 0; i < 16; ++i) aq[i] = ld2(arow + i * 4 + koff);

    // ---- Phase 1: score tiles 0..qt (K tiles via TDM, double-buffered) ----
    auto score_step = [&](int kt, const float* kbuf) {
        v8f c = {};
#pragma unroll
        for (int k0 = 0; k0 < HD; k0 += 4) {
            v2f bb = ld2(kbuf + nl * HD + k0 + koff);  // Kt[k][n]: contiguous in d
            c = __builtin_amdgcn_wmma_f32_16x16x4_f32(false, aq[k0 >> 2], false, bb,
                                                      (short)0, c, false, false);
        }
        int col = kt * 16 + nl;
        float mval = amask[b * S_DIM + col];
#pragma unroll
        for (int r = 0; r < 8; ++r) {
            int row  = rbase + r;
            int qidx = q0 + row;
            float s  = c[r] * 0.125f;                  // 1/sqrt(64)
            bool ok  = (col <= qidx) && (mval != 0.0f);
            sc[row * S_DIM + col] = ok ? s : -__builtin_inff();
        }
    };

    tdm_load_2d(Kb, tile0, HD, 16, HD, S_DIM, HD);
    for (int kt = 0; kt < qt; ++kt) {
        tdm_load_2d(Kb + (size_t)(kt + 1) * 16 * HD,
                    ((kt + 1) & 1) ? tile1 : tile0, HD, 16, HD, S_DIM, HD);
        __builtin_amdgcn_s_wait_tensorcnt(1);
        score_step(kt, (kt & 1) ? tile1 : tile0);
    }
    __builtin_amdgcn_s_wait_tensorcnt(0);
    score_step(qt, (qt & 1) ? tile1 : tile0);
    __syncthreads();

    // ---- Phase 2: softmax, 2 lanes per row ----
    {
        int m    = lane & 15;
        int half = lane >> 4;
        int qidx = q0 + m;
        int kLim = qidx + 1;
        float* srow = sc + m * S_DIM;

        float mx = -__builtin_inff();
        for (int col = half; col < kLim; col += 2) mx = fmaxf(mx, srow[col]);
        mx = fmaxf(mx, __shfl_xor(mx, 16, 32));

        float sum = 0.0f;
        for (int col = half; col < kLim; col += 2) {
            float e = __expf(srow[col] - mx);
            srow[col] = e;
            sum += e;
        }
        sum += __shfl_xor(sum, 16, 32);
        float inv = 1.0f / sum;

        float* grow = attn_out + (((size_t)b * NH + h) * S_DIM + qidx) * S_DIM;
        for (int col = half; col < kLim; col += 2) {
            float p = srow[col] * inv;
            srow[col] = p;
            grow[col] = p;
        }
        int start = kLim + ((kLim ^ half) & 1);  // first col >= kLim, parity `half`
        for (int col = start; col < S_DIM; col += 2) {
            srow[col] = 0.0f;
            grow[col] = 0.0f;
        }
    }
    __syncthreads();

    // ---- Phase 3: ctx = P @ V (V tiles via TDM, 4 persistent accumulators) ----
    v8f acc[4] = {};
    auto ctx_step = [&](int kt, const float* vbuf) {
#pragma unroll
        for (int k0 = 0; k0 < 16; k0 += 4) {
            v2f a = ld2(sc + mrow * S_DIM + kt * 16 + k0 + koff);
#pragma unroll
            for (int nt = 0; nt < 4; ++nt) {
                v2f bb;
                bb.x = vbuf[(k0 + koff)     * HD + nt * 16 + nl];
                bb.y = vbuf[(k0 + koff + 1) * HD + nt * 16 + nl];
                acc[nt] = __builtin_amdgcn_wmma_f32_16x16x4_f32(false, a, false, bb,
                                                                (short)0, acc[nt],
                                                                false, false);
            }
        }
    };

    tdm_load_2d(Vb, tile0, HD, 16, HD, S_DIM, HD);
    for (int kt = 0; kt < qt; ++kt) {
        tdm_load_2d(Vb + (size_t)(kt + 1) * 16 * HD,
                    ((kt + 1) & 1) ? tile1 : tile0, HD, 16, HD, S_DIM, HD);
        __builtin_amdgcn_s_wait_tensorcnt(1);
        ctx_step(kt, (kt & 1) ? tile1 : tile0);
    }
    __builtin_amdgcn_s_wait_tensorcnt(0);
    ctx_step(qt, (qt & 1) ? tile1 : tile0);

#pragma unroll
    for (int nt = 0; nt < 4; ++nt) {
#pragma unroll
        for (int r = 0; r < 8; ++r) {
            int qidx = q0 + rbase + r;
            ctx[((size_t)b * S_DIM + qidx) * HIDDEN + h * HD + nt * 16 + nl]
                = acc[nt][r];
        }
    }
}

// ---------------------------------------------------------------------------
extern "C" void kernel_launch(void* const* d_in, const int* in_sizes, int n_in,
                              void* d_out, int out_size, void* d_ws, size_t ws_size,
                              hipStream_t stream)
{
    const float* hidden = (const float*)d_in[0];
    const float* amask  = (const float*)d_in[1];
    const float* cosT   = (const float*)d_in[2];
    const float* sinT   = (const float*)d_in[3];
    const float* wq     = (const float*)d_in[4];
    const float* wk     = (const float*)d_in[5];
    const float* wv     = (const float*)d_in[6];
    const float* wo     = (const float*)d_in[7];

    float* Qbuf = (float*)d_ws;                                   // 4 Mi floats
    float* Kbuf = Qbuf + (size_t)NH  * B_DIM * S_DIM * HD;        // 1 Mi
    float* Vbuf = Kbuf + (size_t)NKV * B_DIM * S_DIM * HD;        // 1 Mi
    float* ctx  = Vbuf + (size_t)NKV * B_DIM * S_DIM * HD;        // 4 Mi

    float* outp = (float*)d_out;                                  // (B,S,HIDDEN)
    float* attn = outp + (size_t)B_DIM * S_DIM * HIDDEN;          // (B,H,S,S)

    const int M = B_DIM * S_DIM;   // 2048
    dim3 blk(32);
    const size_t gemmLds = 2 * 16 * 16 * sizeof(float);                 // 2 KB
    const size_t attnLds = (16 * S_DIM + 2 * 16 * HD) * sizeof(float);  // 72 KB

    // QKV projections (head-major stores)
    gemm16_wmma<<<dim3((NH  * HD) / 16, M / 16), blk, gemmLds, stream>>>(hidden, wq, Qbuf, M, NH  * HD, HIDDEN, 1);
    gemm16_wmma<<<dim3((NKV * HD) / 16, M / 16), blk, gemmLds, stream>>>(hidden, wk, Kbuf, M, NKV * HD, HIDDEN, 1);
    gemm16_wmma<<<dim3((NKV * HD) / 16, M / 16), blk, gemmLds, stream>>>(hidden, wv, Vbuf, M, NKV * HD, HIDDEN, 1);

    // RoPE on Q and K
    int qThreads = NH  * B_DIM * S_DIM * (HD / 2);
    int kThreads = NKV * B_DIM * S_DIM * (HD / 2);
    rope_kernel<<<(qThreads + 255) / 256, 256, 0, stream>>>(Qbuf, cosT, sinT, NH);
    rope_kernel<<<(kThreads + 255) / 256, 256, 0, stream>>>(Kbuf, cosT, sinT, NKV);

    // Attention core (writes attn output + ctx)
    attn_kernel<<<dim3(S_DIM / 16, NH, B_DIM), blk, attnLds, stream>>>(Qbuf, Kbuf, Vbuf,
                                                                       amask, attn, ctx);

    // Output projection (row-major store)
    gemm16_wmma<<<dim3(HIDDEN / 16, M / 16), blk, gemmLds, stream>>>(ctx, wo, outp, M, HIDDEN, HIDDEN, 0);
}